// AutoCorrelation_56770877718642
// MI455X (gfx1250) — compile-verified
//
#include <hip/hip_runtime.h>

// Problem constants (fixed by the reference: B=16, L=512, H=8, D=64, k=int(ln 512)=6)
#define B_ 16
#define L_ 512
#define H_ 8
#define D_ 64
#define HD_ 512   // H_*D_ (== L_, convenient)
#define K_ 6

typedef float v2f __attribute__((ext_vector_type(2)));
typedef float v8f __attribute__((ext_vector_type(8)));

// ---------------------------------------------------------------------------
// Kernel 1: score[b,l] = (1/4096) * sum_h (sum_d q[b,l,h,d]) * (sum_d k[b,l,h,d])
//
// Rows = flattened (b,l,h): 65536 rows of 64 contiguous fp32.
// One wave handles 16 consecutive rows (a contiguous 4 KB tile) = 2 (b,l) pairs.
// Row sums via V_WMMA_F32_16X16X4_F32 with B = ones, accumulating over 16 K-chunks.
// A-fragment layout (32-bit A 16x4): lanes 0-15 supply K={0,1}, lanes 16-31 K={2,3}
// for row M = lane&15  ->  per lane a float2 at row*64 + 4c + ((lane>=16)?2:0).
// C/D layout: lane<16 holds rows 0..7 in v[0..7]; lane>=16 holds rows 8..15.
// => lane 0 owns all 8 head-sums of pair 2w, lane 16 owns pair 2w+1.
// ---------------------------------------------------------------------------
__global__ void ac_score_kernel(const float* __restrict__ q,
                                const float* __restrict__ k,
                                float* __restrict__ s) {
  const int lane  = threadIdx.x & 31;
  const int gwave = (blockIdx.x * blockDim.x + threadIdx.x) >> 5;   // 0..4095
  const int m     = lane & 15;                                      // tile row
  const int koff  = (lane >> 4) << 1;                               // 0 or 2

  const long rowbase = (long)(gwave * 16 + m) * D_;
  const float* __restrict__ qr = q + rowbase;
  const float* __restrict__ kr = k + rowbase;

  const v2f bones = {1.0f, 1.0f};   // B = ones(4x16): D[m,n] = rowsum(A_m)
  v8f cq = {};
  v8f ck = {};

#pragma unroll
  for (int c = 0; c < 16; ++c) {
    v2f aq = *(const v2f*)(qr + 4 * c + koff);
    v2f ak = *(const v2f*)(kr + 4 * c + koff);
    cq = __builtin_amdgcn_wmma_f32_16x16x4_f32(false, aq, false, bones,
                                               (short)0, cq, false, false);
    ck = __builtin_amdgcn_wmma_f32_16x16x4_f32(false, ak, false, bones,
                                               (short)0, ck, false, false);
  }

  // Per lane: cq[j]/ck[j] = row sums of 8 rows of one (b,l) pair -> sum over heads.
  float dot = 0.0f;
#pragma unroll
  for (int j = 0; j < 8; ++j) dot = fmaf(cq[j], ck[j], dot);

  if (m == 0) {                                  // lanes 0 and 16 write
    const int pair = gwave * 2 + (lane >> 4);    // (b*L + l) flat index
    s[pair] = dot * (1.0f / 4096.0f);            // mean over H*L = 8*512
  }
}

// ---------------------------------------------------------------------------
// Kernel 2: per batch b (one wave): top-6 of s[b,:], softmax, weighted gather
// of values. lax.top_k semantics: descending values, ties -> lower index.
// ---------------------------------------------------------------------------
__global__ void ac_topk_agg_kernel(const float* __restrict__ s,
                                   const float* __restrict__ values,
                                   float* __restrict__ out) {
  const int b    = blockIdx.x;
  const int lane = threadIdx.x & 31;
  const float* __restrict__ sb = s + b * L_;

  // Cache this batch's 512 scores: 16 per lane, strided for coalescing.
  float v[16];
#pragma unroll
  for (int j = 0; j < 16; ++j) v[j] = sb[lane + 32 * j];

  float tv[K_];
  int   ti[K_];

  for (int it = 0; it < K_; ++it) {
    // local argmax (low index wins ties)
    float bv = -3.0e38f;
    int   bi = 0x7fffffff;
#pragma unroll
    for (int j = 0; j < 16; ++j) {
      const int idx = lane + 32 * j;
      if (v[j] > bv || (v[j] == bv && idx < bi)) { bv = v[j]; bi = idx; }
    }
    // wave32 butterfly argmax
#pragma unroll
    for (int off = 16; off >= 1; off >>= 1) {
      const float ov = __shfl_xor(bv, off, 32);
      const int   oi = __shfl_xor(bi, off, 32);
      if (ov > bv || (ov == bv && oi < bi)) { bv = ov; bi = oi; }
    }
    tv[it] = bv;
    ti[it] = bi;
    // knock the winner out (exactly one lane owns it)
#pragma unroll
    for (int j = 0; j < 16; ++j)
      if (lane + 32 * j == bi) v[j] = -3.0e38f;
  }

  // softmax over the 6 (descending) values; tv[0] is the max
  float w[K_];
  float wsum = 0.0f;
#pragma unroll
  for (int i = 0; i < K_; ++i) { w[i] = __expf(tv[i] - tv[0]); wsum += w[i]; }
  const float inv = 1.0f / wsum;

  // out[b, e] = sum_i w_i * values[b, ti[i], e],  e over H*D = 512 (contig)
  const float* __restrict__ vb = values + (long)b * L_ * HD_;
  float* __restrict__ ob = out + b * HD_;
  for (int e = lane; e < HD_; e += 32) {
    float acc = 0.0f;
#pragma unroll
    for (int i = 0; i < K_; ++i)
      acc = fmaf(w[i] * inv, vb[(long)ti[i] * HD_ + e], acc);
    ob[e] = acc;
  }
}

// ---------------------------------------------------------------------------
extern "C" void kernel_launch(void* const* d_in, const int* in_sizes, int n_in,
                              void* d_out, int out_size, void* d_ws, size_t ws_size,
                              hipStream_t stream) {
  const float* q      = (const float*)d_in[0];   // [16,512,8,64] f32
  const float* k      = (const float*)d_in[1];   // [16,512,8,64] f32
  const float* values = (const float*)d_in[2];   // [16,512,8,64] f32
  float* s   = (float*)d_ws;                     // [16,512] score map (32 KB)
  float* out = (float*)d_out;                    // [16,8,64] f32

  // 65536 rows / 16 rows-per-wave = 4096 waves; 8 waves (256 thr) per block.
  ac_score_kernel<<<512, 256, 0, stream>>>(q, k, s);
  ac_topk_agg_kernel<<<B_, 32, 0, stream>>>(s, values, out);
}